// KeyFormer_317827580173
// MI455X (gfx1250) — compile-verified
//
#include <hip/hip_runtime.h>
#include <hip/hip_bf16.h>

// KeyFormer attention for MI455X (gfx1250), fp32 WMMA path.
// M=16, N=4096, D=128, H=32, P=8192, L=P+M=8208.

typedef __attribute__((ext_vector_type(2))) float v2f;
typedef __attribute__((ext_vector_type(8))) float v8f;

#define MM 16
#define NN 4096
#define DD 128
#define HH 32
#define PP 8192
#define LL 8208  // P + M

static __device__ __forceinline__ v8f wmma_f32(v2f a, v2f b, v8f c) {
  // D = A(16x4 f32) * B(4x16 f32) + C(16x16 f32)
  return __builtin_amdgcn_wmma_f32_16x16x4_f32(
      /*neg_a=*/false, a, /*neg_b=*/false, b,
      /*c_mod=*/(short)0, c, /*reuse_a=*/false, /*reuse_b=*/false);
}

// ---------------------------------------------------------------------------
// K1: QKV projection.  grid = (64 col-groups of 64, 3 matrices), block = 32.
// Each wave computes a 16x64 output tile: K-loop over 4096 in steps of 4,
// reusing the A (X) fragment across 4 N-tiles.
// ---------------------------------------------------------------------------
__global__ void kf_qkv_gemm(const float* __restrict__ X,
                            const float* __restrict__ Wq,
                            const float* __restrict__ Wk,
                            const float* __restrict__ Wv,
                            float* __restrict__ qo,
                            float* __restrict__ ko,
                            float* __restrict__ vo) {
  const int lane  = threadIdx.x;          // 0..31
  const int cb    = blockIdx.x * 64;      // output column base
  const int which = blockIdx.y;           // 0=q 1=k 2=v
  const float* __restrict__ W = (which == 0) ? Wq : (which == 1) ? Wk : Wv;
  float* __restrict__ O       = (which == 0) ? qo : (which == 1) ? ko : vo;

  const int n     = lane & 15;            // A row (m) / B,C column (n)
  const int khalf = (lane >> 4) * 2;      // K sub-offset: 0 (lo half) or 2 (hi half)

  v8f c0 = {}, c1 = {}, c2 = {}, c3 = {};
  const float* __restrict__ xrow = X + (size_t)n * NN;

  for (int k0 = 0; k0 < NN; k0 += 4) {
    // A fragment: lane(m=n) holds X[m][k0+khalf .. +1]
    v2f a = *(const v2f*)(xrow + k0 + khalf);
    // B fragments: lane holds W[k0+khalf][col], W[k0+khalf+1][col]
    const float* wr0 = W + (size_t)(k0 + khalf) * NN + cb + n;
    const float* wr1 = wr0 + NN;
    v2f b0 = {wr0[0],  wr1[0]};
    v2f b1 = {wr0[16], wr1[16]};
    v2f b2 = {wr0[32], wr1[32]};
    v2f b3 = {wr0[48], wr1[48]};
    c0 = wmma_f32(a, b0, c0);
    c1 = wmma_f32(a, b1, c1);
    c2 = wmma_f32(a, b2, c2);
    c3 = wmma_f32(a, b3, c3);
  }
  // C/D layout: VGPR r -> (M = r + 8*(lane>=16), N = lane&15)
  const int mbase = (lane >> 4) * 8;
#pragma unroll
  for (int r = 0; r < 8; ++r) {
    const size_t row = (size_t)(r + mbase) * NN + cb + n;
    O[row +  0] = c0[r];
    O[row + 16] = c1[r];
    O[row + 32] = c2[r];
    O[row + 48] = c3[r];
  }
}

// ---------------------------------------------------------------------------
// K2: scores[h][m][s] = q_h[m,:] . K_h[s,:]   (K = cache_K ++ fresh k)
// grid = (32 heads, 513 seq-tiles), block = 32.  32 WMMAs over D=128.
// ---------------------------------------------------------------------------
__global__ void kf_scores(const float* __restrict__ q,
                          const float* __restrict__ kws,
                          const float* __restrict__ cacheK,
                          float* __restrict__ scores) {
  const int lane  = threadIdx.x;
  const int h     = blockIdx.x;
  const int kb    = blockIdx.y * 16;      // key tile base
  const int n     = lane & 15;
  const int khalf = (lane >> 4) * 2;

  const int row = kb + n;                 // this lane's key row (B column n)
  const float* __restrict__ krow =
      (row < PP) ? cacheK + ((size_t)h * PP + row) * DD
                 : kws + (size_t)(row - PP) * NN + (size_t)h * DD;
  const float* __restrict__ qrow = q + (size_t)n * NN + (size_t)h * DD;

  v8f c = {};
  for (int d0 = 0; d0 < DD; d0 += 4) {
    v2f a = *(const v2f*)(qrow + d0 + khalf);  // A: q[m][d0+khalf..+1]
    v2f b = *(const v2f*)(krow + d0 + khalf);  // B: K[row][d0+khalf..+1]
    c = wmma_f32(a, b, c);
  }
  const int mbase = (lane >> 4) * 8;
#pragma unroll
  for (int r = 0; r < 8; ++r) {
    scores[((size_t)h * MM + (r + mbase)) * LL + kb + n] = c[r];
  }
}

// ---------------------------------------------------------------------------
// K3: per-row softmax (clean + noise-perturbed).  grid = (32, 16), block=256.
// Writes perturb_out to d_out, overwrites scores row with normalized weights.
// ---------------------------------------------------------------------------
__global__ void kf_softmax(const float* __restrict__ noise,
                           float* __restrict__ scores,
                           float* __restrict__ perturb_out) {
  const int h = blockIdx.x, m = blockIdx.y, tid = threadIdx.x;
  const size_t base = ((size_t)h * MM + m) * LL;
  float* __restrict__ srow = scores + base;
  const float* __restrict__ nrow = noise + base;
  float* __restrict__ prow = perturb_out + base;

  __shared__ float red[256];
  const float inv15 = 1.0f / 1.5f;

  float ms = -3.0e38f, mp = -3.0e38f;
  for (int i = tid; i < LL; i += 256) {
    const float s = srow[i];
    const float p = (s + nrow[i]) * inv15;
    ms = fmaxf(ms, s);
    mp = fmaxf(mp, p);
  }
  red[tid] = ms; __syncthreads();
  for (int o = 128; o > 0; o >>= 1) { if (tid < o) red[tid] = fmaxf(red[tid], red[tid + o]); __syncthreads(); }
  ms = red[0]; __syncthreads();
  red[tid] = mp; __syncthreads();
  for (int o = 128; o > 0; o >>= 1) { if (tid < o) red[tid] = fmaxf(red[tid], red[tid + o]); __syncthreads(); }
  mp = red[0]; __syncthreads();

  float ss = 0.0f, sp = 0.0f;
  for (int i = tid; i < LL; i += 256) {
    const float s = srow[i];
    const float p = (s + nrow[i]) * inv15;
    ss += expf(s - ms);
    sp += expf(p - mp);
  }
  red[tid] = ss; __syncthreads();
  for (int o = 128; o > 0; o >>= 1) { if (tid < o) red[tid] += red[tid + o]; __syncthreads(); }
  ss = red[0]; __syncthreads();
  red[tid] = sp; __syncthreads();
  for (int o = 128; o > 0; o >>= 1) { if (tid < o) red[tid] += red[tid + o]; __syncthreads(); }
  sp = red[0]; __syncthreads();

  const float iss = 1.0f / ss, isp = 1.0f / sp;
  for (int i = tid; i < LL; i += 256) {
    const float s = srow[i];
    const float p = (s + nrow[i]) * inv15;
    srow[i] = expf(s - ms) * iss;          // normalized attention weight
    prow[i] = expf(p - mp) * isp;          // keyformer perturbed softmax
  }
}

// ---------------------------------------------------------------------------
// K4: output[h][m][d] = weights[h][m,:] @ V_h[:, d]   (V = cache_V ++ fresh v)
// grid = (32 heads, 8 D-tiles), block = 32.  2052-step K-loop over seq.
// ---------------------------------------------------------------------------
__global__ void kf_out_gemm(const float* __restrict__ weights,
                            const float* __restrict__ vws,
                            const float* __restrict__ cacheV,
                            float* __restrict__ out) {
  const int lane  = threadIdx.x;
  const int h     = blockIdx.x;
  const int nb    = blockIdx.y * 16;      // d-column base
  const int n     = lane & 15;
  const int khalf = (lane >> 4) * 2;

  const float* __restrict__ wrow = weights + ((size_t)h * MM + n) * LL;
  const size_t cvh = (size_t)h * PP;

  v8f c = {};
  for (int k0 = 0; k0 < LL; k0 += 4) {
    v2f a = *(const v2f*)(wrow + k0 + khalf);      // A: w[m][k0+khalf..+1]
    const int r0 = k0 + khalf;
    const float* v0 = (r0 < PP) ? cacheV + (cvh + r0) * DD
                                : vws + (size_t)(r0 - PP) * NN + (size_t)h * DD;
    const float* v1 = (r0 + 1 < PP) ? cacheV + (cvh + r0 + 1) * DD
                                    : vws + (size_t)(r0 + 1 - PP) * NN + (size_t)h * DD;
    v2f b = {v0[nb + n], v1[nb + n]};              // B: V[r][nb+n]
    c = wmma_f32(a, b, c);
  }
  const int mbase = (lane >> 4) * 8;
#pragma unroll
  for (int r = 0; r < 8; ++r) {
    out[(size_t)(r + mbase) * NN + (size_t)h * DD + nb + n] = c[r];
  }
}

// ---------------------------------------------------------------------------
extern "C" void kernel_launch(void* const* d_in, const int* in_sizes, int n_in,
                              void* d_out, int out_size, void* d_ws, size_t ws_size,
                              hipStream_t stream) {
  (void)in_sizes; (void)n_in; (void)out_size; (void)ws_size;
  const float* X      = (const float*)d_in[0];
  const float* Wq     = (const float*)d_in[1];
  const float* Wk     = (const float*)d_in[2];
  const float* Wv     = (const float*)d_in[3];
  const float* noise  = (const float*)d_in[4];
  const float* cacheK = (const float*)d_in[5];
  const float* cacheV = (const float*)d_in[6];
  float* out = (float*)d_out;                    // [0,65536): output, then perturb_out

  float* ws     = (float*)d_ws;
  float* q      = ws;                            // 16*4096
  float* k      = ws + 65536;                    // 16*4096
  float* v      = ws + 131072;                   // 16*4096
  float* scores = ws + 196608;                   // 32*16*8208

  kf_qkv_gemm<<<dim3(64, 3), 32, 0, stream>>>(X, Wq, Wk, Wv, q, k, v);
  kf_scores  <<<dim3(HH, LL / 16), 32, 0, stream>>>(q, k, cacheK, scores);
  kf_softmax <<<dim3(HH, MM), 256, 0, stream>>>(noise, scores, out + 65536);
  kf_out_gemm<<<dim3(HH, DD / 16), 32, 0, stream>>>(scores, v, cacheV, out);
}